// KVCache_47339129536539
// MI455X (gfx1250) — compile-verified
//
#include <hip/hip_runtime.h>
#include <cstdint>

// KV-cache decode-step expand for MI455X (gfx1250).
// Pure bandwidth op (0 FLOPs): read 128 MiB (K+V layer cache), write 512 MiB
// (n_rep=4 GQA broadcast of both) -> ~29 us roofline at 23.3 TB/s.
// Uses the CDNA5 async global<->LDS DMA path: each 512 B head row is
// async-DMA'd into LDS once (ASYNCcnt), then the n_rep replicated head copies
// are async-stored straight from LDS (fire-and-forget, no VGPR staging).
// Non-temporal hints on both sides: zero reuse, output alone exceeds L2.

#define BSZ       4
#define MAX_SEQ   4096
#define KVH       8
#define HD        128
#define WAVES_PER_BLOCK 8
#define ROWS_PER_WAVE   8

__global__ __launch_bounds__(256) void kv_expand_async(
    const float* __restrict__ xk,       // [BSZ, L, KVH, HD]
    const float* __restrict__ xv,       // [BSZ, L, KVH, HD]
    const float* __restrict__ k_cache,  // [LAYERS, BSZ, MAX_SEQ, KVH, HD]
    const float* __restrict__ v_cache,  // [LAYERS, BSZ, MAX_SEQ, KVH, HD]
    const int* __restrict__ layer_idx_p,
    const int* __restrict__ cur_pos_p,
    const int* __restrict__ n_rep_p,
    float* __restrict__ out,            // [2, BSZ, MAX_SEQ, KVH*n_rep, HD]
    int L)
{
    // 8 waves x 8 rows x 512 B = 32 KB LDS per block (of 320 KB/WGP).
    __shared__ __align__(16) float lds_buf[WAVES_PER_BLOCK * ROWS_PER_WAVE * HD];

    const int lane = threadIdx.x & 31;
    const int wave = threadIdx.x >> 5;
    const int wave_global = blockIdx.x * WAVES_PER_BLOCK + wave;
    const int row0 = wave_global * ROWS_PER_WAVE;

    constexpr int ROWS_TOTAL = 2 * BSZ * MAX_SEQ * KVH; // 262144 = 2^18
    if (row0 >= ROWS_TOTAL) return;                      // wave-uniform guard

    // Uniform scalars from device memory (single-element int arrays).
    const int layer = *layer_idx_p;
    const int cur   = *cur_pos_p;
    const int n_rep = *n_rep_p;
    const int H     = KVH * n_rep;

    // Wave-private LDS region; generic shared address low 32 bits are the
    // wave-relative LDS byte offset.
    const unsigned lds_wave =
        (unsigned)(uintptr_t)(&lds_buf[wave * ROWS_PER_WAVE * HD]);

    float* dst_base[ROWS_PER_WAVE];

    // Phase 1: issue all async loads (8 rows = 8 x 512 B in flight per wave).
    #pragma unroll
    for (int j = 0; j < ROWS_PER_WAVE; ++j) {
        const int row   = row0 + j;
        const int kvsel = row >> 17;                        // / (BSZ*MAX_SEQ*KVH)
        const int rem   = row & ((BSZ * MAX_SEQ * KVH) - 1);
        const int b     = rem >> 15;                        // / (MAX_SEQ*KVH)
        const int rem2  = rem & ((MAX_SEQ * KVH) - 1);
        const int s     = rem2 >> 3;                        // / KVH
        const int kvh   = rem2 & (KVH - 1);

        // Read-side override of the cache insertion (do NOT mutate inputs).
        const float* src;
        if (s >= cur && s < cur + L) {
            const float* x = kvsel ? xv : xk;
            src = x + (((size_t)b * L + (size_t)(s - cur)) * KVH + kvh) * HD;
        } else {
            const float* c = kvsel ? v_cache : k_cache;
            src = c + ((((size_t)layer * BSZ + b) * MAX_SEQ + s) * KVH + kvh) * HD;
        }

        const unsigned lds_a = lds_wave + (unsigned)(j * HD * 4)
                                        + (unsigned)(lane * 16);
        const uint64_t ga = (uint64_t)(uintptr_t)(src + lane * 4);
        asm volatile("global_load_async_to_lds_b128 %0, %1, off th:TH_LOAD_NT"
                     :: "v"(lds_a), "v"(ga) : "memory");

        dst_base[j] = out
            + ((((size_t)kvsel * BSZ + b) * MAX_SEQ + s) * H
               + (size_t)kvh * n_rep) * HD
            + lane * 4;
    }

    // One wait per 8-row batch: async loads must land in LDS before the
    // async stores read it (load/store LDS sides are not ordered otherwise).
    asm volatile("s_wait_asynccnt 0" ::: "memory");

    // Phase 2: fan out n_rep replicated head rows per source row, straight
    // from LDS (fire-and-forget; s_endpgm's implicit wait-idle covers them).
    #pragma unroll
    for (int j = 0; j < ROWS_PER_WAVE; ++j) {
        const unsigned lds_a = lds_wave + (unsigned)(j * HD * 4)
                                        + (unsigned)(lane * 16);
        float* d = dst_base[j];
        for (int r = 0; r < n_rep; ++r) {
            const uint64_t gd = (uint64_t)(uintptr_t)(d + (size_t)r * HD);
            asm volatile("global_store_async_from_lds_b128 %0, %1, off th:TH_STORE_NT"
                         :: "v"(gd), "v"(lds_a) : "memory");
        }
    }
}

extern "C" void kernel_launch(void* const* d_in, const int* in_sizes, int n_in,
                              void* d_out, int out_size, void* d_ws, size_t ws_size,
                              hipStream_t stream) {
    const float* xk      = (const float*)d_in[0];
    const float* xv      = (const float*)d_in[1];
    const float* k_cache = (const float*)d_in[2];
    const float* v_cache = (const float*)d_in[3];
    const int* layer_idx = (const int*)d_in[4];
    const int* cur_pos   = (const int*)d_in[5];
    const int* n_rep     = (const int*)d_in[6];
    float* out           = (float*)d_out;

    const int L = in_sizes[0] / (BSZ * KVH * HD);   // insert length from xk shape

    constexpr int ROWS_TOTAL = 2 * BSZ * MAX_SEQ * KVH;          // 262144
    const int waves  = (ROWS_TOTAL + ROWS_PER_WAVE - 1) / ROWS_PER_WAVE; // 32768
    const int blocks = (waves + WAVES_PER_BLOCK - 1) / WAVES_PER_BLOCK;  // 4096

    kv_expand_async<<<blocks, WAVES_PER_BLOCK * 32, 0, stream>>>(
        xk, xv, k_cache, v_cache, layer_idx, cur_pos, n_rep, out, L);
}